// HighFreqSuppressionLoss_60035052863716
// MI455X (gfx1250) — compile-verified
//
#include <hip/hip_runtime.h>
#include <math.h>

// ---------------------------------------------------------------------------
// HighFreqSuppressionLoss on MI455X (gfx1250, wave32)
//
//   gray = (0.299 R + 0.587 G + 0.114 B + 1) / 2
//   F    = FFT2(gray)                       (512 x 512, per image)
//   psd  = mean over high-freq bins of log10(|F|^2 + 1e-10)
//   loss = mean_b |psd_gen[b] - psd_tgt[b]|
//
// Row pass  : wave-per-row radix-2 DIT FFT in LDS (luma fused into load).
// Col pass  : DFT512 x (512 x 32 slab) complex GEMM on V_WMMA_F32_16X16X4_F32,
//             DFT matrix pre-packed in the WMMA A-operand lane/vgpr layout,
//             slab staged in 128 KB LDS, masked log10-PSD reduction fused
//             in-register (spectrum never returns to HBM).
//             F32 WMMA has no A/B NEG modifier (NEG = {CNeg,0,0}), so the
//             -Aim*Bim term uses a VALU-negated copy of the 2-dword A operand.
// fftshift  : handled by wrapped-index mask: bin masked-out iff
//             (kr<64 || kr>=448) && (kc<64 || kc>=448).  #high bins = 245760.
// ---------------------------------------------------------------------------

typedef float v2f __attribute__((ext_vector_type(2)));
typedef float v8f __attribute__((ext_vector_type(8)));

#define PI_F 3.14159265358979323846f

#define CHUNK   16          // images per chunk (64 total -> 4 chunks)
#define NCHUNK  4

// Workspace layout (bytes)
#define A_FLOATS (32 * 128 * 2 * 64)     // ktile x kchunk x {re,im} x (2*lane32)
#define A_OFF    0                        // 2 MB packed DFT matrix
#define PART_OFF (A_FLOATS * 4)           // 64 images x 16 slabs partial sums
#define SPEC_OFF (PART_OFF + 4096)        // CHUNK x 512 x 512 complex = 32 MB

// ---------------------------------------------------------------------------
// Pack W[kr][r] = exp(-2*pi*i*kr*r/512) into WMMA A-operand order:
//   A 16x4 f32 layout (ISA 7.12.2): element (m,kk) -> lane = m + 16*(kk>>1),
//   vgpr = kk&1.  Flat: ((t*128+q)*2+comp)*64 + 2*lane + vgpr  (b64/lane load).
// ---------------------------------------------------------------------------
__global__ void init_tw_kernel(float* __restrict__ apack) {
  int f = blockIdx.x * blockDim.x + threadIdx.x;
  if (f >= A_FLOATS) return;
  int v    = f & 1;
  int l    = (f >> 1) & 31;
  int rest = f >> 6;
  int comp = rest & 1;
  int q    = (rest >> 1) & 127;
  int t    = rest >> 8;
  int kr = t * 16 + (l & 15);
  int r  = q * 4 + v + 2 * (l >> 4);
  float ang = -2.0f * PI_F * (float)((kr * r) & 511) * (1.0f / 512.0f);
  apack[f] = comp ? sinf(ang) : cosf(ang);
}

// ---------------------------------------------------------------------------
// Row FFT: one wave per image row.  8 waves / 256-thread block, 32 KB LDS.
// ---------------------------------------------------------------------------
__global__ __launch_bounds__(256) void row_fft_kernel(
    const float* __restrict__ gen, const float* __restrict__ tgt,
    float* __restrict__ spec, int chunk) {
  __shared__ float sre[8][512];
  __shared__ float sim[8][512];
  const int wave = threadIdx.x >> 5;
  const int lane = threadIdx.x & 31;
  const int rowGlobal  = blockIdx.x * 8 + wave;      // [0, CHUNK*512)
  const int imgInChunk = rowGlobal >> 9;
  const int r          = rowGlobal & 511;
  const int g          = chunk * CHUNK + imgInChunk; // global image id [0,64)
  const float* src = (g < 32) ? gen : tgt;
  const int    b   = (g < 32) ? g : (g - 32);
  const float* rp = src + (((size_t)b * 3 + 0) * 512 + r) * 512;
  const float* gp = rp + 512 * 512;
  const float* bp = gp + 512 * 512;

  // luma + bit-reversed scatter into LDS (DIT: bit-rev in, natural out)
  #pragma unroll
  for (int j = 0; j < 16; ++j) {
    int p = lane + 32 * j;                           // coalesced 128B loads
    float gray = (0.299f * rp[p] + 0.587f * gp[p] + 0.114f * bp[p] + 1.0f) * 0.5f;
    int q = __brev((unsigned)p) >> 23;               // 9-bit reverse
    sre[wave][q] = gray;
    sim[wave][q] = 0.0f;
  }
  __syncthreads();

  for (int s = 0; s < 9; ++s) {
    const int half = 1 << s;
    #pragma unroll
    for (int k = 0; k < 8; ++k) {
      int bf  = lane + 32 * k;                       // 256 disjoint butterflies
      int pos = bf & (half - 1);
      int i0  = ((bf >> s) << (s + 1)) | pos;
      int i1  = i0 + half;
      float sw, cw;
      __sincosf(-PI_F * (float)pos / (float)half, &sw, &cw);
      float a1r = sre[wave][i1], a1i = sim[wave][i1];
      float tr = cw * a1r - sw * a1i;
      float ti = cw * a1i + sw * a1r;
      float a0r = sre[wave][i0], a0i = sim[wave][i0];
      sre[wave][i1] = a0r - tr;
      sim[wave][i1] = a0i - ti;
      sre[wave][i0] = a0r + tr;
      sim[wave][i0] = a0i + ti;
    }
    __syncthreads();
  }

  float2* o = (float2*)spec + ((size_t)imgInChunk * 512 + r) * 512;
  #pragma unroll
  for (int j = 0; j < 16; ++j) {
    int p = lane + 32 * j;                           // coalesced 256B stores
    o[p] = make_float2(sre[wave][p], sim[wave][p]);
  }
}

// ---------------------------------------------------------------------------
// Column transform as complex GEMM on WMMA f32 16x16x4 + fused masked PSD.
// Block = 16 waves (512 thr), one (image, 32-col slab).  128 KB dynamic LDS.
// Wave w computes k-tiles {w, w+16}, both 16-col halves fused in the K loop.
// ---------------------------------------------------------------------------
static __device__ __forceinline__ v8f wmma_f32(v2f a, v2f bm, v8f c) {
  return __builtin_amdgcn_wmma_f32_16x16x4_f32(false, a, false, bm, (short)0, c,
                                               false, false);
}

__global__ __launch_bounds__(512) void col_dft_psd_kernel(
    const float* __restrict__ spec, const float* __restrict__ apack,
    float* __restrict__ part, int chunk) {
  extern __shared__ float2 lds2[];                   // 512 x 32 complex
  __shared__ float wacc[32];
  const int imgInChunk = blockIdx.x >> 4;
  const int slab       = blockIdx.x & 15;
  const int g          = chunk * CHUNK + imgInChunk;
  const float2* xs = (const float2*)spec + (size_t)imgInChunk * 512 * 512;

  // stage slab: each wave iteration reads one contiguous 256B row segment
  for (int it = 0; it < 32; ++it) {
    int idx = it * 512 + threadIdx.x;
    int r = idx >> 5, c = idx & 31;
    lds2[r * 32 + c] = xs[(size_t)r * 512 + slab * 32 + c];
  }
  __syncthreads();

  const int wave = threadIdx.x >> 5;
  const int lane = threadIdx.x & 31;
  const int lh   = lane >> 4;                        // lane half (K-stripe)
  const int lm   = lane & 15;                        // N / M-within-tile index

  for (int ti = 0; ti < 2; ++ti) {
    const int tt = wave + 16 * ti;                   // k-tile [0,32)
    v8f dre0 = {0,0,0,0,0,0,0,0};
    v8f dim0 = dre0, dre1 = dre0, dim1 = dre0;
    const float2* ap = (const float2*)(apack + (size_t)tt * (128 * 2 * 64));

    for (int q = 0; q < 128; ++q) {                  // K = 512, 4 per chunk
      float2 a_re = ap[(q * 2 + 0) * 32 + lane];     // coalesced b64: vgpr0,1
      float2 a_im = ap[(q * 2 + 1) * 32 + lane];
      v2f Are  = { a_re.x,  a_re.y};
      v2f Aim  = { a_im.x,  a_im.y};
      v2f AimN = {-a_im.x, -a_im.y};                 // f32 WMMA has no neg_a
      // B 4x16: element (kk,n) -> lane = n + 16*(kk>>1), vgpr = kk&1
      int r0 = q * 4 + 2 * lh;
      float2 x00 = lds2[(r0    ) * 32 + lm];
      float2 x01 = lds2[(r0 + 1) * 32 + lm];
      float2 x10 = lds2[(r0    ) * 32 + 16 + lm];
      float2 x11 = lds2[(r0 + 1) * 32 + 16 + lm];
      v2f Bre0 = {x00.x, x01.x}, Bim0 = {x00.y, x01.y};
      v2f Bre1 = {x10.x, x11.x}, Bim1 = {x10.y, x11.y};
      // complex MAC: Dre += Ar*Br - Ai*Bi ; Dim += Ar*Bi + Ai*Br
      dre0 = wmma_f32(Are,  Bre0, dre0);
      dre0 = wmma_f32(AimN, Bim0, dre0);
      dim0 = wmma_f32(Are,  Bim0, dim0);
      dim0 = wmma_f32(Aim,  Bre0, dim0);
      dre1 = wmma_f32(Are,  Bre1, dre1);
      dre1 = wmma_f32(AimN, Bim1, dre1);
      dim1 = wmma_f32(Are,  Bim1, dim1);
      dim1 = wmma_f32(Aim,  Bre1, dim1);
    }

    // fused masked log10-PSD over the 16x16 D tiles (C/D layout: vgpr j ->
    // M = j + 8*(lane>>4), N = lane&15)
    float acc = 0.0f;
    const int  kc0   = slab * 32 + lm;
    const int  kc1   = kc0 + 16;
    const bool lowc0 = (kc0 < 64) || (kc0 >= 448);
    const bool lowc1 = (kc1 < 64) || (kc1 >= 448);
    #pragma unroll
    for (int j = 0; j < 8; ++j) {
      int  kr   = tt * 16 + j + 8 * lh;
      bool lowr = (kr < 64) || (kr >= 448);
      float p0 = dre0[j] * dre0[j] + dim0[j] * dim0[j];
      float p1 = dre1[j] * dre1[j] + dim1[j] * dim1[j];
      if (!(lowr && lowc0)) acc += log10f(p0 + 1e-10f);
      if (!(lowr && lowc1)) acc += log10f(p1 + 1e-10f);
    }
    for (int o = 16; o > 0; o >>= 1) acc += __shfl_xor(acc, o, 32);
    if (lane == 0) wacc[wave * 2 + ti] = acc;
  }
  __syncthreads();

  if (threadIdx.x == 0) {                            // deterministic (no atomics)
    float s = 0.0f;
    #pragma unroll
    for (int i = 0; i < 32; ++i) s += wacc[i];
    part[(size_t)g * 16 + slab] = s;
  }
}

// ---------------------------------------------------------------------------
// loss = (1/32) * sum_b | (Sg[b] - St[b]) / 245760 |
// ---------------------------------------------------------------------------
__global__ void final_loss_kernel(const float* __restrict__ part,
                                  float* __restrict__ out) {
  int b = threadIdx.x;                               // 32 threads
  float sg = 0.0f, st = 0.0f;
  #pragma unroll
  for (int s = 0; s < 16; ++s) {
    sg += part[(size_t)b * 16 + s];
    st += part[(size_t)(b + 32) * 16 + s];
  }
  float d = fabsf(sg - st) * (1.0f / 245760.0f);
  for (int o = 16; o > 0; o >>= 1) d += __shfl_xor(d, o, 32);
  if (b == 0) out[0] = d * (1.0f / 32.0f);
}

// ---------------------------------------------------------------------------
extern "C" void kernel_launch(void* const* d_in, const int* in_sizes, int n_in,
                              void* d_out, int out_size, void* d_ws,
                              size_t ws_size, hipStream_t stream) {
  const float* gen = (const float*)d_in[0];
  const float* tgt = (const float*)d_in[1];
  float* out = (float*)d_out;
  char*  ws  = (char*)d_ws;
  float* apack = (float*)(ws + A_OFF);
  float* part  = (float*)(ws + PART_OFF);
  float* spec  = (float*)(ws + SPEC_OFF);

  init_tw_kernel<<<A_FLOATS / 256, 256, 0, stream>>>(apack);
  for (int chunk = 0; chunk < NCHUNK; ++chunk) {
    row_fft_kernel<<<CHUNK * 512 / 8, 256, 0, stream>>>(gen, tgt, spec, chunk);
    col_dft_psd_kernel<<<CHUNK * 16, 512, 512 * 32 * sizeof(float2), stream>>>(
        spec, apack, part, chunk);
  }
  final_loss_kernel<<<1, 32, 0, stream>>>(part, out);
}